// GraphSAGE_NET_58969900974220
// MI455X (gfx1250) — compile-verified
//
#include <hip/hip_runtime.h>

typedef float v2f __attribute__((ext_vector_type(2)));
typedef float v8f __attribute__((ext_vector_type(8)));

#define DF 128   // feature / hidden dim (both 128 in this net)

// ---------------------------------------------------------------- zero fill
__global__ void zero_f32(float* __restrict__ p, int n) {
    int i = blockIdx.x * blockDim.x + threadIdx.x;
    if (i < n) p[i] = 0.0f;
}

// ------------------------------------------------- edge scatter (mean prep)
// One wave32 per edge: lane l handles features [4l, 4l+4). Gather x[src] with
// a b128 load, scatter-add into agg[dst] with f32 atomics (L2-resident).
__global__ void sage_scatter(const float* __restrict__ feat,
                             const int*   __restrict__ ei,
                             float* __restrict__ agg,
                             float* __restrict__ cnt,
                             int n_edges, int with_cnt) {
    const int lane   = threadIdx.x & 31;
    const int wave   = blockIdx.x * (blockDim.x >> 5) + (threadIdx.x >> 5);
    const int nwaves = gridDim.x * (blockDim.x >> 5);
    for (int e = wave; e < n_edges; e += nwaves) {
        const int src = ei[e];
        const int dst = ei[n_edges + e];
        const float4 v = *(const float4*)(feat + (size_t)src * DF + lane * 4);
        float* a = agg + (size_t)dst * DF + lane * 4;
        atomicAdd(a + 0, v.x);
        atomicAdd(a + 1, v.y);
        atomicAdd(a + 2, v.z);
        atomicAdd(a + 3, v.w);
        if (with_cnt && lane == 0) atomicAdd(cnt + dst, 1.0f);
    }
}

// ----------------------------------------------- agg -> mean (in place)
__global__ void sage_mean(float* __restrict__ agg, const float* __restrict__ cnt,
                          int n_elems) {
    int i = blockIdx.x * blockDim.x + threadIdx.x;
    if (i < n_elems) {
        float c = fmaxf(cnt[i >> 7], 1.0f);   // DF == 128 -> row = i >> 7
        agg[i] = agg[i] / c;
    }
}

// ---------------------------------------------------------------- fused GEMM
// out = relu( Amean @ Wl^T + bl + Askip @ Wr^T ), all [n,128] x [128,128].
// Block = 256 threads = 8 waves; block owns 16 node rows, wave w owns output
// cols [16w, 16w+16). Both matmuls feed one v8f accumulator via
// V_WMMA_F32_16X16X4_F32 (fp32 in / fp32 acc, matches reference precision).
//
// fp32 WMMA operand layouts (ISA 7.12.2), lane l = threadIdx&31:
//   A 16x4 : A[l%16][ 2*(l/16) + v ]         v = reg 0..1
//   B 4x16 : B[ 2*(l/16) + v ][ l%16 ]  = W[col0 + l%16][ k ]  (W is [out,in])
//   C 16x16: reg r -> row r + 8*(l/16), col l%16
__global__ void __launch_bounds__(256)
sage_gemm_relu(const float* __restrict__ Amean, const float* __restrict__ Askip,
               const float* __restrict__ Wl, const float* __restrict__ bl,
               const float* __restrict__ Wr, float* __restrict__ out,
               int n_nodes) {
    const int lane  = threadIdx.x & 31;
    const int wave  = threadIdx.x >> 5;
    const int row0  = blockIdx.x << 4;     // 16-node tile
    const int col0  = wave << 4;           // 16-col tile (8 waves cover 128)
    const int m     = lane & 15;
    const int khalf = (lane >> 4) << 1;    // 0 or 2

    int arow = row0 + m;
    if (arow >= n_nodes) arow = n_nodes - 1;   // clamp (no tail for N=50000)

    const float* Am = Amean + (size_t)arow * DF;
    const float* Ax = Askip + (size_t)arow * DF;
    const float* Bl = Wl + (size_t)(col0 + m) * DF;
    const float* Br = Wr + (size_t)(col0 + m) * DF;

    v8f acc = {0.f, 0.f, 0.f, 0.f, 0.f, 0.f, 0.f, 0.f};

#pragma unroll
    for (int k = 0; k < DF; k += 4) {
        v2f a = *(const v2f*)(Am + k + khalf);
        v2f b = *(const v2f*)(Bl + k + khalf);
        acc = __builtin_amdgcn_wmma_f32_16x16x4_f32(false, a, false, b,
                                                    (short)0, acc, false, false);
    }
#pragma unroll
    for (int k = 0; k < DF; k += 4) {
        v2f a = *(const v2f*)(Ax + k + khalf);
        v2f b = *(const v2f*)(Br + k + khalf);
        acc = __builtin_amdgcn_wmma_f32_16x16x4_f32(false, a, false, b,
                                                    (short)0, acc, false, false);
    }

    const float bias  = bl[col0 + m];
    const int   rbase = row0 + ((lane >> 4) << 3);
#pragma unroll
    for (int r = 0; r < 8; ++r) {
        int row = rbase + r;
        float v = acc[r] + bias;
        v = v > 0.0f ? v : 0.0f;
        if (row < n_nodes) out[(size_t)row * DF + (col0 + m)] = v;
    }
}

// ---------------------------------------------------------------- dispatcher
extern "C" void kernel_launch(void* const* d_in, const int* in_sizes, int n_in,
                              void* d_out, int out_size, void* d_ws, size_t ws_size,
                              hipStream_t stream) {
    const float* x   = (const float*)d_in[0];
    const int*   ei  = (const int*)d_in[1];   // JAX x64 off -> int32 [2,E]
    const float* W1l = (const float*)d_in[2];
    const float* b1l = (const float*)d_in[3];
    const float* W1r = (const float*)d_in[4];
    const float* W2l = (const float*)d_in[5];
    const float* b2l = (const float*)d_in[6];
    const float* W2r = (const float*)d_in[7];
    float* out = (float*)d_out;

    const int N = in_sizes[0] / DF;
    const int E = in_sizes[1] / 2;
    const int nf = N * DF;

    float* agg = (float*)d_ws;               // [N,128]
    float* h1  = agg + (size_t)N * DF;       // [N,128]
    float* cnt = h1  + (size_t)N * DF;       // [N]

    dim3 b256(256);
    dim3 gZeroF((nf + 255) / 256);
    dim3 gZeroC((N + 255) / 256);
    dim3 gScat(2048);
    dim3 gGemm((N + 15) / 16);

    // ---- layer 1: h1 = relu(mean(x) @ W1l^T + b1 + x @ W1r^T)
    zero_f32<<<gZeroF, b256, 0, stream>>>(agg, nf);
    zero_f32<<<gZeroC, b256, 0, stream>>>(cnt, N);
    sage_scatter<<<gScat, b256, 0, stream>>>(x, ei, agg, cnt, E, 1);
    sage_mean<<<gZeroF, b256, 0, stream>>>(agg, cnt, nf);
    sage_gemm_relu<<<gGemm, b256, 0, stream>>>(agg, x, W1l, b1l, W1r, h1, N);

    // ---- layer 2: out = relu(mean(h1) @ W2l^T + b2 + h1 @ W2r^T)
    zero_f32<<<gZeroF, b256, 0, stream>>>(agg, nf);
    sage_scatter<<<gScat, b256, 0, stream>>>(h1, ei, agg, cnt, E, 0);
    sage_mean<<<gZeroF, b256, 0, stream>>>(agg, cnt, nf);
    sage_gemm_relu<<<gGemm, b256, 0, stream>>>(agg, h1, W2l, b2l, W2r, out, N);
}